// ChamferLoss_84043920048708
// MI455X (gfx1250) — compile-verified
//
#include <hip/hip_runtime.h>

typedef __attribute__((ext_vector_type(2))) float v2f;
typedef __attribute__((ext_vector_type(8))) float v8f;

#define KITTI_SCALE   80.0f
#define ROWS_PER_WAVE 32      // two 16-row WMMA tiles per wave
#define THREADS       256     // 8 waves
#define ROWS_PER_BLK  (ROWS_PER_WAVE * (THREADS / 32))   // 256
#define CHUNK         512     // Q points staged per LDS round
#define FLT_BIG       3.402823466e38f

__global__ void chamfer_zero_out(float* out) { out[0] = 0.0f; }

// One direction of chamfer: for each row point p in P, min over all column
// points q in Q of |80p - 80q|^2; row-mins summed * outScale into out[0].
//
// K=4 fully used:  A_row = (-2x, -2y, -2z, 1) * 80-scale on coords
//                  B_col = ( x,   y,   z, |q|^2 )   (scaled)
//  =>  A.B = |q|^2 - 2 p.q ;  C[j] = |p_row|^2 (loop-invariant)
//  =>  D = |p|^2 + |q|^2 - 2 p.q  in a single V_WMMA_F32_16X16X4_F32.
// Clamp-to-zero is monotone, so it is applied once after the min-reduction.
__global__ __launch_bounds__(THREADS)
void chamfer_rowmin_wmma(const float* __restrict__ P,   // [B, N, 3] row side
                         const float* __restrict__ Q,   // [B, M, 3] col side
                         float* __restrict__ out,
                         int N, int M, float outScale)
{
    __shared__ float4 sP[ROWS_PER_BLK];   // scaled {x,y,z,|p|^2}
    __shared__ float4 sQ[CHUNK];          // scaled {x,y,z,|q|^2}

    const int nRowBlocks = N / ROWS_PER_BLK;
    const int b       = blockIdx.x / nRowBlocks;
    const int rowBase = (blockIdx.x % nRowBlocks) * ROWS_PER_BLK;

    const float* __restrict__ Pb = P + (size_t)b * (size_t)N * 3u;
    const float* __restrict__ Qb = Q + (size_t)b * (size_t)M * 3u;

    const int tid = threadIdx.x;

    // ---- stage this block's row points (scaled) + norms into LDS ----
    {
        const float* p = Pb + (size_t)(rowBase + tid) * 3u;
        float x = p[0] * KITTI_SCALE;
        float y = p[1] * KITTI_SCALE;
        float z = p[2] * KITTI_SCALE;
        sP[tid] = make_float4(x, y, z, x * x + y * y + z * z);
    }
    __syncthreads();

    const int lane = tid & 31;
    const int wave = tid >> 5;
    const int half = lane >> 4;   // 0: lanes 0-15 carry K0,K1 / 1: lanes 16-31 carry K2,K3
    const int lid  = lane & 15;
    const int wrow = wave * ROWS_PER_WAVE;

    // Two A tiles (16x4 f32 layout), rows pre-scaled by -2, K3 = 1.
    float4 p0 = sP[wrow + lid];
    float4 p1 = sP[wrow + 16 + lid];
    v2f A0, A1;
    A0.x = (half ? p0.z : p0.x) * -2.0f;
    A0.y =  half ? 1.0f : (p0.y * -2.0f);
    A1.x = (half ? p1.z : p1.x) * -2.0f;
    A1.y =  half ? 1.0f : (p1.y * -2.0f);

    // Loop-invariant C tiles: |p|^2 for the 8 rows each C/D VGPR maps to.
    v8f C0, C1;
#pragma unroll
    for (int j = 0; j < 8; ++j) {
        C0[j] = sP[wrow + j + half * 8].w;
        C1[j] = sP[wrow + 16 + j + half * 8].w;
    }

    float rmin0[8], rmin1[8];
#pragma unroll
    for (int j = 0; j < 8; ++j) { rmin0[j] = FLT_BIG; rmin1[j] = FLT_BIG; }

    for (int m0 = 0; m0 < M; m0 += CHUNK) {
        __syncthreads();
        // ---- stage CHUNK column points (scaled) + norms into LDS ----
        for (int p = tid; p < CHUNK; p += THREADS) {
            const float* q = Qb + (size_t)(m0 + p) * 3u;
            float x = q[0] * KITTI_SCALE;
            float y = q[1] * KITTI_SCALE;
            float z = q[2] * KITTI_SCALE;
            sQ[p] = make_float4(x, y, z, x * x + y * y + z * z);
        }
        if (m0 + CHUNK < M)
            __builtin_prefetch((const void*)(Qb + (size_t)(m0 + CHUNK) * 3u + tid), 0, 1);
        __syncthreads();

        // ---- two 16x16 distance tiles per LDS load ----
#pragma unroll 4
        for (int t = 0; t < CHUNK / 16; ++t) {
            float4 q = sQ[t * 16 + lid];
            v2f Bm;
            Bm.x = half ? q.z : q.x;   // K0 / K2
            Bm.y = half ? q.w : q.y;   // K1 / K3 (= |q|^2)

            v8f D0 = __builtin_amdgcn_wmma_f32_16x16x4_f32(
                false, A0, false, Bm, (short)0, C0, false, false);
#pragma unroll
            for (int j = 0; j < 8; ++j) rmin0[j] = fminf(rmin0[j], D0[j]);

            v8f D1 = __builtin_amdgcn_wmma_f32_16x16x4_f32(
                false, A1, false, Bm, (short)0, C1, false, false);
#pragma unroll
            for (int j = 0; j < 8; ++j) rmin1[j] = fminf(rmin1[j], D1[j]);
        }
    }

    // ---- row-min spread over 16 lanes of each half: xor-reduce ----
#pragma unroll
    for (int off = 1; off < 16; off <<= 1) {
#pragma unroll
        for (int j = 0; j < 8; ++j) {
            rmin0[j] = fminf(rmin0[j], __shfl_xor(rmin0[j], off, 32));
            rmin1[j] = fminf(rmin1[j], __shfl_xor(rmin1[j], off, 32));
        }
    }
    // deferred clamp (monotone) + sum of the 16+16 row minima
    float s = 0.0f;
#pragma unroll
    for (int j = 0; j < 8; ++j) {
        s += fmaxf(rmin0[j], 0.0f);
        s += fmaxf(rmin1[j], 0.0f);
    }
    // low half holds rows 0-7 of each tile, high half rows 8-15: combine
    s += __shfl_xor(s, 16, 32);

    if (lane == 0) atomicAdd(out, s * outScale);
}

extern "C" void kernel_launch(void* const* d_in, const int* in_sizes, int n_in,
                              void* d_out, int out_size, void* d_ws, size_t ws_size,
                              hipStream_t stream) {
    const float* pred = (const float*)d_in[0];   // [B, N, 3]
    const float* gt   = (const float*)d_in[1];   // [B, M, 3]
    float* out = (float*)d_out;

    const int B = 2;
    const int N = in_sizes[0] / (3 * B);
    const int M = in_sizes[1] / (3 * B);

    chamfer_zero_out<<<1, 1, 0, stream>>>(out);

    dim3 block(THREADS);
    dim3 gridP(B * (N / ROWS_PER_BLK));
    dim3 gridQ(B * (M / ROWS_PER_BLK));
    // pred -> gt direction: mean over N rows, then mean over B
    chamfer_rowmin_wmma<<<gridP, block, 0, stream>>>(pred, gt, out, N, M,
                                                     1.0f / (float)(N * B));
    // gt -> pred direction: mean over M rows, then mean over B
    chamfer_rowmin_wmma<<<gridQ, block, 0, stream>>>(gt, pred, out, M, N,
                                                     1.0f / (float)(M * B));
}